// GRUModel_20418274525278
// MI455X (gfx1250) — compile-verified
//
#include <hip/hip_runtime.h>
#include <math.h>

// Problem dimensions (fixed by the reference).
#define T_DIM 128
#define B_DIM 32
#define V_DIM 10000
#define H_DIM 1024

typedef __attribute__((ext_vector_type(2))) float v2f;
typedef __attribute__((ext_vector_type(8))) float v8f;
typedef __attribute__((ext_vector_type(4))) int   v4i;

// Detect the gfx1250 async global->LDS builtins (ASYNCcnt-tracked).
#if defined(__has_builtin)
#if __has_builtin(__builtin_amdgcn_global_load_async_to_lds_b128)
#define HAS_ASYNC_LDS 1
#endif
#endif
#ifndef HAS_ASYNC_LDS
#define HAS_ASYNC_LDS 0
#endif

// ---------------------------------------------------------------------------
// CDNA5 fp32 WMMA: D(16x16,f32) = A(16x4,f32) * B(4x16,f32) + C
// ---------------------------------------------------------------------------
__device__ __forceinline__ v8f wmma_f32(v2f a, v2f b, v8f c) {
  return __builtin_amdgcn_wmma_f32_16x16x4_f32(false, a, false, b,
                                               (short)0, c, false, false);
}

__device__ __forceinline__ float sigmoidf_(float x) {
  return 1.0f / (1.0f + __expf(-x));
}

// Copy 16 aligned bytes global->LDS. Valid quads use the async path
// (global_load_async_to_lds_b128, tracked by ASYNCcnt); invalid quads are
// zero-filled with a plain ds_store (DScnt-tracked, covered by the barrier).
__device__ __forceinline__ void async_copy16(float* ldsDst, const float* gSrc,
                                             bool valid) {
#if HAS_ASYNC_LDS
  if (valid) {
    float* g = const_cast<float*>(gSrc);
    __builtin_amdgcn_global_load_async_to_lds_b128(
        (__attribute__((address_space(1))) v4i*)g,
        (__attribute__((address_space(3))) v4i*)ldsDst, 0, 0);
  } else {
    *(float4*)ldsDst = make_float4(0.f, 0.f, 0.f, 0.f);
  }
#else
  float4 v = make_float4(0.f, 0.f, 0.f, 0.f);
  if (valid) v = *(const float4*)gSrc;
  *(float4*)ldsDst = v;
#endif
}

__device__ __forceinline__ void wait_async_all() {
#if HAS_ASYNC_LDS
#if __has_builtin(__builtin_amdgcn_s_wait_asynccnt)
  __builtin_amdgcn_s_wait_asynccnt(0);
#else
  asm volatile("s_wait_asynccnt 0" ::: "memory");
#endif
#endif
}

// ---------------------------------------------------------------------------
// Fused triple projection: C_g = A @ W_g + b_g  for g in {0,1,2}
// A:[M,K], W_g:[K,N], C_g:[M,N].  BM=64, BN=64, KT=32, 256 threads.
// Double-buffered async global->LDS staging; A tile shared by all 3 gates.
// LDS strides padded to multiples of 4 floats (16B-aligned async writes),
// A-stride 36 -> conflict-free fragment reads (36r mod 64 distinct, r<16).
// ---------------------------------------------------------------------------
__global__ __launch_bounds__(256)
void proj3_kernel(const float* __restrict__ A,
                  const float* __restrict__ W0, const float* __restrict__ W1,
                  const float* __restrict__ W2,
                  const float* __restrict__ b0, const float* __restrict__ b1,
                  const float* __restrict__ b2,
                  float* __restrict__ C0, float* __restrict__ C1,
                  float* __restrict__ C2,
                  int M, int N, int K)
{
  __shared__ float As[2][64][36];
  __shared__ float Bs[2][3][32][68];

  const int tid  = threadIdx.x;
  const int lane = tid & 31;
  const int wv   = tid >> 5;
  const int m0   = blockIdx.y * 64;
  const int n0   = blockIdx.x * 64;

  const int mr   = wv & 3;            // 16-row band
  const int c0   = (wv >> 2) * 2;     // two adjacent 16-col tiles
  const int arow = lane & 15;
  const int kb   = (lane >> 4) << 1;  // 0 or 2

  v8f acc[3][2] = {};
  const int nk = (K + 31) >> 5;

  auto stage = [&](int ck, int buf) {
    const int kc = ck << 5;
#pragma unroll
    for (int i = 0; i < 2; ++i) {     // A: 512 quads, 2/thread
      int idx = tid + i * 256;
      int r = idx >> 3, q = (idx & 7) << 2;
      int gk = kc + q;
      async_copy16(&As[buf][r][q], A + (size_t)(m0 + r) * K + gk, gk < K);
    }
#pragma unroll
    for (int i = 0; i < 6; ++i) {     // B: 1536 quads, 6/thread
      int idx = tid + i * 256;
      int g = idx >> 9;
      int rem = idx & 511;
      int r = rem >> 4, cq = (rem & 15) << 2;
      int gk = kc + r;
      const float* Wg = (g == 0) ? W0 : ((g == 1) ? W1 : W2);
      async_copy16(&Bs[buf][g][r][cq], Wg + (size_t)gk * N + n0 + cq, gk < K);
    }
  };

  stage(0, 0);
  for (int ck = 0; ck < nk; ++ck) {
    const int cur = ck & 1;
    wait_async_all();
    __syncthreads();                       // buffer `cur` fully staged
    if (ck + 1 < nk) stage(ck + 1, cur ^ 1);  // overlap next stage w/ compute

#pragma unroll
    for (int kk = 0; kk < 32; kk += 4) {
      v2f a;
      a.x = As[cur][mr * 16 + arow][kk + kb];
      a.y = As[cur][mr * 16 + arow][kk + kb + 1];
#pragma unroll
      for (int g = 0; g < 3; ++g) {
        v2f bb0, bb1;
        bb0.x = Bs[cur][g][kk + kb    ][c0 * 16 + arow];
        bb0.y = Bs[cur][g][kk + kb + 1][c0 * 16 + arow];
        bb1.x = Bs[cur][g][kk + kb    ][(c0 + 1) * 16 + arow];
        bb1.y = Bs[cur][g][kk + kb + 1][(c0 + 1) * 16 + arow];
        acc[g][0] = wmma_f32(a, bb0, acc[g][0]);
        acc[g][1] = wmma_f32(a, bb1, acc[g][1]);
      }
    }
    __syncthreads();                       // done reading `cur` before reuse
  }

  const int em = (lane >> 4) << 3;
#pragma unroll
  for (int g = 0; g < 3; ++g) {
    const float* bg = (g == 0) ? b0 : ((g == 1) ? b1 : b2);
    float* Cg       = (g == 0) ? C0 : ((g == 1) ? C1 : C2);
#pragma unroll
    for (int ci = 0; ci < 2; ++ci) {
      int gn = n0 + (c0 + ci) * 16 + arow;
      float bv = bg[gn];
      v8f av = ci ? acc[g][1] : acc[g][0];
#pragma unroll
      for (int v = 0; v < 8; ++v) {
        int gm = m0 + mr * 16 + v + em;
        Cg[(size_t)gm * N + gn] = av[v] + bv;
      }
    }
  }
}

// ---------------------------------------------------------------------------
// Generic GEMM + bias: C = A @ B + bias.  BM=64, BN=64, KT=32, 256 threads.
// Double-buffered async staging; handles N tail (N=10000) with guards.
// ---------------------------------------------------------------------------
__global__ __launch_bounds__(256)
void gemm_bias_kernel(const float* __restrict__ A, const float* __restrict__ Bm,
                      const float* __restrict__ bias, float* __restrict__ C,
                      int M, int N, int K)
{
  __shared__ float As[2][64][36];
  __shared__ float Bs[2][32][68];

  const int tid  = threadIdx.x;
  const int lane = tid & 31;
  const int wv   = tid >> 5;
  const int m0   = blockIdx.y * 64;
  const int n0   = blockIdx.x * 64;

  const int mr   = wv & 3;
  const int c0   = (wv >> 2) * 2;
  const int arow = lane & 15;
  const int kb   = (lane >> 4) << 1;

  v8f acc0 = {}, acc1 = {};
  const int nk = (K + 31) >> 5;

  auto stage = [&](int ck, int buf) {
    const int kc = ck << 5;
#pragma unroll
    for (int i = 0; i < 2; ++i) {
      int idx = tid + i * 256;
      int r = idx >> 3, q = (idx & 7) << 2;
      int gk = kc + q, gm = m0 + r;
      async_copy16(&As[buf][r][q], A + (size_t)gm * K + gk, gk < K && gm < M);
    }
#pragma unroll
    for (int i = 0; i < 2; ++i) {
      int idx = tid + i * 256;
      int r = idx >> 4, cq = (idx & 15) << 2;
      int gk = kc + r, gn = n0 + cq;
      async_copy16(&Bs[buf][r][cq], Bm + (size_t)gk * N + gn, gk < K && gn < N);
    }
  };

  stage(0, 0);
  for (int ck = 0; ck < nk; ++ck) {
    const int cur = ck & 1;
    wait_async_all();
    __syncthreads();
    if (ck + 1 < nk) stage(ck + 1, cur ^ 1);

#pragma unroll
    for (int kk = 0; kk < 32; kk += 4) {
      v2f a;
      a.x = As[cur][mr * 16 + arow][kk + kb];
      a.y = As[cur][mr * 16 + arow][kk + kb + 1];
      v2f bb0, bb1;
      bb0.x = Bs[cur][kk + kb    ][c0 * 16 + arow];
      bb0.y = Bs[cur][kk + kb + 1][c0 * 16 + arow];
      bb1.x = Bs[cur][kk + kb    ][(c0 + 1) * 16 + arow];
      bb1.y = Bs[cur][kk + kb + 1][(c0 + 1) * 16 + arow];
      acc0 = wmma_f32(a, bb0, acc0);
      acc1 = wmma_f32(a, bb1, acc1);
    }
    __syncthreads();
  }

  const int em = (lane >> 4) << 3;
#pragma unroll
  for (int ci = 0; ci < 2; ++ci) {
    int gn = n0 + (c0 + ci) * 16 + arow;
    if (gn < N) {
      float bv = bias[gn];
      v8f av = ci ? acc1 : acc0;
#pragma unroll
      for (int v = 0; v < 8; ++v) {
        int gm = m0 + mr * 16 + v + em;
        if (gm < M) C[(size_t)gm * N + gn] = av[v] + bv;
      }
    }
  }
}

// ---------------------------------------------------------------------------
// GRU step, phase 1: Z = sigmoid(Xz + H@Whz); RH = sigmoid(Xr + H@Whr) * H
// M = B_DIM = 32 rows, N = H_DIM, K = H_DIM.  grid.x = H_DIM/64.
// ---------------------------------------------------------------------------
__global__ __launch_bounds__(256)
void gru_zr_kernel(const float* __restrict__ Hprev,
                   const float* __restrict__ Whz, const float* __restrict__ Whr,
                   const float* __restrict__ Xz_t, const float* __restrict__ Xr_t,
                   float* __restrict__ Zbuf, float* __restrict__ RHbuf)
{
  __shared__ float As[32][33];
  __shared__ float Bs[2][32][65];

  const int tid  = threadIdx.x;
  const int lane = tid & 31;
  const int wv   = tid >> 5;
  const int n0   = blockIdx.x * 64;

  const int g    = wv & 1;            // 0 = Z gate, 1 = R gate
  const int mr   = (wv >> 1) & 1;     // row tile (batch 0..15 / 16..31)
  const int c0   = (wv >> 2) * 2;     // column tile pair
  const int arow = lane & 15;
  const int kb   = (lane >> 4) << 1;

  v8f acc0 = {}, acc1 = {};

  for (int ck = 0; ck < H_DIM / 32; ++ck) {
    const int kc = ck << 5;
    {  // A tile (Hprev): 32x32 = 256 float4, 1 per thread
      int r = tid >> 3, q = (tid & 7) << 2;
      float4 va = *(const float4*)(Hprev + (size_t)r * H_DIM + kc + q);
      As[r][q + 0] = va.x; As[r][q + 1] = va.y;
      As[r][q + 2] = va.z; As[r][q + 3] = va.w;
    }
#pragma unroll
    for (int i = 0; i < 4; ++i) {  // 2 B tiles: 1024 float4, 4 per thread
      int idx = tid + i * 256;
      int bg = idx >> 9;
      int rem = idx & 511;
      int r = rem >> 4, cq = (rem & 15) << 2;
      const float* Wp = bg ? Whr : Whz;
      float4 vb = *(const float4*)(Wp + (size_t)(kc + r) * H_DIM + n0 + cq);
      Bs[bg][r][cq + 0] = vb.x; Bs[bg][r][cq + 1] = vb.y;
      Bs[bg][r][cq + 2] = vb.z; Bs[bg][r][cq + 3] = vb.w;
    }
    __syncthreads();
#pragma unroll
    for (int kk = 0; kk < 32; kk += 4) {
      v2f a;
      a.x = As[mr * 16 + arow][kk + kb];
      a.y = As[mr * 16 + arow][kk + kb + 1];
      v2f bb0, bb1;
      bb0.x = Bs[g][kk + kb    ][c0 * 16 + arow];
      bb0.y = Bs[g][kk + kb + 1][c0 * 16 + arow];
      bb1.x = Bs[g][kk + kb    ][(c0 + 1) * 16 + arow];
      bb1.y = Bs[g][kk + kb + 1][(c0 + 1) * 16 + arow];
      acc0 = wmma_f32(a, bb0, acc0);
      acc1 = wmma_f32(a, bb1, acc1);
    }
    __syncthreads();
  }

  const int em = (lane >> 4) << 3;
#pragma unroll
  for (int ci = 0; ci < 2; ++ci) {
    int gn = n0 + (c0 + ci) * 16 + arow;
    v8f av = ci ? acc1 : acc0;
#pragma unroll
    for (int v = 0; v < 8; ++v) {
      int b = mr * 16 + v + em;  // batch row 0..31
      size_t off = (size_t)b * H_DIM + gn;
      if (g == 0) {
        Zbuf[off] = sigmoidf_(av[v] + Xz_t[off]);
      } else {
        float r = sigmoidf_(av[v] + Xr_t[off]);
        RHbuf[off] = r * Hprev[off];
      }
    }
  }
}

// ---------------------------------------------------------------------------
// GRU step, phase 2: Ht = tanh(Xh + RH@Whh); Hn = Z*H + (1-Z)*Ht
// ---------------------------------------------------------------------------
__global__ __launch_bounds__(256)
void gru_h_kernel(const float* __restrict__ RHbuf, const float* __restrict__ Whh,
                  const float* __restrict__ Xh_t, const float* __restrict__ Zbuf,
                  const float* __restrict__ Hprev, float* __restrict__ Hnew)
{
  __shared__ float As[32][33];
  __shared__ float Bs[32][65];

  const int tid  = threadIdx.x;
  const int lane = tid & 31;
  const int wv   = tid >> 5;
  const int n0   = blockIdx.x * 64;

  const int mr   = wv & 1;        // row tile
  const int ct   = wv >> 1;       // column tile 0..3
  const int arow = lane & 15;
  const int kb   = (lane >> 4) << 1;

  v8f acc = {};

  for (int ck = 0; ck < H_DIM / 32; ++ck) {
    const int kc = ck << 5;
    {  // A tile (RH): 256 float4
      int r = tid >> 3, q = (tid & 7) << 2;
      float4 va = *(const float4*)(RHbuf + (size_t)r * H_DIM + kc + q);
      As[r][q + 0] = va.x; As[r][q + 1] = va.y;
      As[r][q + 2] = va.z; As[r][q + 3] = va.w;
    }
#pragma unroll
    for (int i = 0; i < 2; ++i) {  // B tile: 512 float4
      int idx = tid + i * 256;
      int r = idx >> 4, cq = (idx & 15) << 2;
      float4 vb = *(const float4*)(Whh + (size_t)(kc + r) * H_DIM + n0 + cq);
      Bs[r][cq + 0] = vb.x; Bs[r][cq + 1] = vb.y;
      Bs[r][cq + 2] = vb.z; Bs[r][cq + 3] = vb.w;
    }
    __syncthreads();
#pragma unroll
    for (int kk = 0; kk < 32; kk += 4) {
      v2f a;
      a.x = As[mr * 16 + arow][kk + kb];
      a.y = As[mr * 16 + arow][kk + kb + 1];
      v2f bb;
      bb.x = Bs[kk + kb    ][ct * 16 + arow];
      bb.y = Bs[kk + kb + 1][ct * 16 + arow];
      acc = wmma_f32(a, bb, acc);
    }
    __syncthreads();
  }

  const int em = (lane >> 4) << 3;
  const int gn = n0 + ct * 16 + arow;
#pragma unroll
  for (int v = 0; v < 8; ++v) {
    int b = mr * 16 + v + em;
    size_t off = (size_t)b * H_DIM + gn;
    float ht = tanhf(acc[v] + Xh_t[off]);
    float z  = Zbuf[off];
    Hnew[off] = z * Hprev[off] + (1.0f - z) * ht;
  }
}

// ---------------------------------------------------------------------------
// Host-side orchestration (graph-capture safe).
// ---------------------------------------------------------------------------
extern "C" void kernel_launch(void* const* d_in, const int* in_sizes, int n_in,
                              void* d_out, int out_size, void* d_ws, size_t ws_size,
                              hipStream_t stream) {
  const float* inputs = (const float*)d_in[0];   // [T,B,V]
  const float* H0     = (const float*)d_in[1];   // [B,H]
  const float* W_xz   = (const float*)d_in[2];
  const float* W_hz   = (const float*)d_in[3];
  const float* b_z    = (const float*)d_in[4];
  const float* W_xr   = (const float*)d_in[5];
  const float* W_hr   = (const float*)d_in[6];
  const float* b_r    = (const float*)d_in[7];
  const float* W_xh   = (const float*)d_in[8];
  const float* W_hh   = (const float*)d_in[9];
  const float* b_h    = (const float*)d_in[10];
  const float* W_hq   = (const float*)d_in[11];
  const float* b_q    = (const float*)d_in[12];
  float* out = (float*)d_out;

  const int M = T_DIM * B_DIM;           // 4096
  const size_t MH = (size_t)M * H_DIM;
  const size_t BH = (size_t)B_DIM * H_DIM;

  float* ws = (float*)d_ws;
  float* Xz = ws;              // [T*B, H]
  float* Xr = Xz + MH;
  float* Xh = Xr + MH;
  float* Hs = Xh + MH;         // [T, B, H]
  float* Zb = Hs + MH;         // [B, H]
  float* RH = Zb + BH;         // [B, H]

  dim3 blk(256);

  // 1) Batched input projections (fused: one pass over `inputs`).
  proj3_kernel<<<dim3(H_DIM / 64, M / 64), blk, 0, stream>>>(
      inputs, W_xz, W_xr, W_xh, b_z, b_r, b_h, Xz, Xr, Xh, M, H_DIM, V_DIM);

  // 2) Sequential recurrence: two dependent GEMM phases per timestep.
  for (int t = 0; t < T_DIM; ++t) {
    const float* Hprev = (t == 0) ? H0 : (Hs + (size_t)(t - 1) * BH);
    const float* Xz_t = Xz + (size_t)t * BH;
    const float* Xr_t = Xr + (size_t)t * BH;
    const float* Xh_t = Xh + (size_t)t * BH;
    float* Hs_t = Hs + (size_t)t * BH;

    gru_zr_kernel<<<dim3(H_DIM / 64), blk, 0, stream>>>(
        Hprev, W_hz, W_hr, Xz_t, Xr_t, Zb, RH);
    gru_h_kernel<<<dim3(H_DIM / 64), blk, 0, stream>>>(
        RH, W_hh, Xh_t, Zb, Hprev, Hs_t);
  }

  // 3) Output projection: [T*B,H] @ [H,V] + b_q -> outputs.
  gemm_bias_kernel<<<dim3((V_DIM + 63) / 64, M / 64), blk, 0, stream>>>(
      Hs, W_hq, b_q, out, M, V_DIM, H_DIM);

  // 4) H_last = Hs[T-1]  (appended after outputs in d_out).
  (void)hipMemcpyAsync(out + (size_t)T_DIM * B_DIM * V_DIM,
                       Hs + (size_t)(T_DIM - 1) * BH,
                       BH * sizeof(float), hipMemcpyDeviceToDevice, stream);
}